// Differentiate_78580721647776
// MI455X (gfx1250) — compile-verified
//
#include <hip/hip_runtime.h>

// Problem constants (match reference): B=32, C=3, L=512, E=512
#define BB 32
#define CC 3
#define LL 512
#define EE 512
#define BCT (BB * CC)

typedef float v2f __attribute__((ext_vector_type(2)));
typedef float v8f __attribute__((ext_vector_type(8)));

#if defined(__gfx1250__) && __has_builtin(__builtin_amdgcn_wmma_f32_16x16x4_f32)
#define USE_WMMA_F32 1
#else
#define USE_WMMA_F32 0
#endif

// ---------------------------------------------------------------------------
// K1: per (b,c,f) projections: T0,T1 = t{0,1}@Tw ; S = (x_last@Qw*T0+Qb)*(u0@Kw*T1+Kb)
// ---------------------------------------------------------------------------
__global__ __launch_bounds__(128) void proj_kernel(
    const float* __restrict__ x0, const float* __restrict__ u0,
    const float* __restrict__ t0, const float* __restrict__ t1,
    const float* __restrict__ Qw, const float* __restrict__ Qb,
    const float* __restrict__ Kw, const float* __restrict__ Kb,
    const float* __restrict__ Tw,
    float* __restrict__ T0, float* __restrict__ T1, float* __restrict__ S)
{
    int f  = blockIdx.x * 128 + threadIdx.x;
    int bc = blockIdx.y;
    int b  = bc / CC, c = bc % CC;
    const float* tw  = Tw + (size_t)c * EE * EE + f;
    const float* qw  = Qw + (size_t)c * EE * EE + f;
    const float* kw  = Kw + (size_t)c * EE * EE + f;
    const float* t0p = t0 + (size_t)b * EE;
    const float* t1p = t1 + (size_t)b * EE;
    const float* xp  = x0 + ((size_t)bc * LL + (LL - 1)) * EE;  // x0[b,c,L-1,:]
    const float* up  = u0 + (size_t)bc * EE;

    float aT0 = 0.f, aT1 = 0.f, aQ = 0.f, aK = 0.f;
    for (int e = 0; e < EE; ++e) {
        float wv = tw[(size_t)e * EE];
        aT0 = fmaf(t0p[e], wv, aT0);
        aT1 = fmaf(t1p[e], wv, aT1);
        aQ  = fmaf(xp[e], qw[(size_t)e * EE], aQ);
        aK  = fmaf(up[e], kw[(size_t)e * EE], aK);
    }
    size_t o = (size_t)bc * EE + f;
    T0[o] = aT0;
    T1[o] = aT1;
    float Qv = aQ * aT0 + Qb[c * EE + f];
    float Kv = aK * aT1 + Kb[c * EE + f];
    S[o] = Qv * Kv;
}

// ---------------------------------------------------------------------------
// K2: per (b,c): w = sum(T0*T1)/sqrt(sum T0^2 + sum T1^2); cb = sum(S*V_bias)
// ---------------------------------------------------------------------------
__global__ __launch_bounds__(256) void wcb_kernel(
    const float* __restrict__ T0, const float* __restrict__ T1,
    const float* __restrict__ S, const float* __restrict__ Vb,
    float* __restrict__ W, float* __restrict__ CB)
{
    __shared__ float r0[256], r1[256], r2[256], r3[256];
    int bc = blockIdx.x, t = threadIdx.x, c = bc % CC;
    size_t base = (size_t)bc * EE;
    float n = 0.f, d0 = 0.f, d1 = 0.f, cb = 0.f;
    for (int e = t; e < EE; e += 256) {
        float a = T0[base + e], bv = T1[base + e];
        n += a * bv; d0 += a * a; d1 += bv * bv;
        cb += S[base + e] * Vb[c * EE + e];
    }
    r0[t] = n; r1[t] = d0; r2[t] = d1; r3[t] = cb;
    __syncthreads();
    for (int s2 = 128; s2 > 0; s2 >>= 1) {
        if (t < s2) { r0[t] += r0[t + s2]; r1[t] += r1[t + s2];
                      r2[t] += r2[t + s2]; r3[t] += r3[t + s2]; }
        __syncthreads();
    }
    if (t == 0) {
        W[bc]  = r0[0] / sqrtf(r1[0] + r2[0]);
        CB[bc] = r3[0];
    }
}

// ---------------------------------------------------------------------------
// K3: g[b,c,e] = sum_f (S*T1)[f] * Vw[c,e,f]  (one wave per e-row, lanes over f)
// ---------------------------------------------------------------------------
__global__ __launch_bounds__(256) void g_kernel(
    const float* __restrict__ S, const float* __restrict__ T1,
    const float* __restrict__ Vw, float* __restrict__ G)
{
    __shared__ float sv[EE];
    int bc = blockIdx.x, c = bc % CC, t = threadIdx.x;
    size_t base = (size_t)bc * EE;
    for (int f = t; f < EE; f += 256) sv[f] = S[base + f] * T1[base + f];
    __syncthreads();
    int wave = t >> 5, lane = t & 31;
    int e = blockIdx.y * 8 + wave;
    const float* vw = Vw + ((size_t)c * EE + e) * EE;
    float acc = 0.f;
    for (int f = lane; f < EE; f += 32) acc = fmaf(sv[f], vw[f], acc);
    for (int off = 16; off > 0; off >>= 1) acc += __shfl_down(acc, off, 32);
    if (lane == 0) G[base + e] = acc;
}

// ---------------------------------------------------------------------------
// K4: logits[b,c,l] = x0[b,c,l,:] . g[b,c,:] + cb[b,c]  (one wave per l)
// ---------------------------------------------------------------------------
__global__ __launch_bounds__(256) void logit_kernel(
    const float* __restrict__ x0, const float* __restrict__ G,
    const float* __restrict__ CB, float* __restrict__ QKV)
{
    __shared__ float gb[EE];
    int bc = blockIdx.x, t = threadIdx.x;
    size_t base = (size_t)bc * EE;
    for (int e = t; e < EE; e += 256) gb[e] = G[base + e];
    __syncthreads();
    int wave = t >> 5, lane = t & 31;
    int l = blockIdx.y * 8 + wave;
    const float* xr = x0 + ((size_t)bc * LL + l) * EE;
    float acc = 0.f;
    for (int e = lane; e < EE; e += 32) acc = fmaf(xr[e], gb[e], acc);
    for (int off = 16; off > 0; off >>= 1) acc += __shfl_down(acc, off, 32);
    if (lane == 0) QKV[(size_t)bc * LL + l] = acc + CB[bc];
}

// ---------------------------------------------------------------------------
// K5: in-place softmax over L per (b,c)
// ---------------------------------------------------------------------------
__global__ __launch_bounds__(256) void softmax_kernel(float* __restrict__ QKV)
{
    __shared__ float red[256];
    int bc = blockIdx.x, t = threadIdx.x;
    size_t base = (size_t)bc * LL;
    float v0 = QKV[base + t], v1 = QKV[base + t + 256];
    red[t] = fmaxf(v0, v1);
    __syncthreads();
    for (int s2 = 128; s2 > 0; s2 >>= 1) {
        if (t < s2) red[t] = fmaxf(red[t], red[t + s2]);
        __syncthreads();
    }
    float M = red[0];
    __syncthreads();
    float e0 = expf(v0 - M), e1 = expf(v1 - M);
    red[t] = e0 + e1;
    __syncthreads();
    for (int s2 = 128; s2 > 0; s2 >>= 1) {
        if (t < s2) red[t] += red[t + s2];
        __syncthreads();
    }
    float inv = 1.f / red[0];
    QKV[base + t]       = e0 * inv;
    QKV[base + t + 256] = e1 * inv;
}

// ---------------------------------------------------------------------------
// K6: out0 = A1 - A0 (pure streaming, float4). l<L-1: A0[l+1]-A0[0]-A0[l];
//     l==L-1: QKV - A0[0]. A0 (100MB) fits the 192MB L2, re-reads are cheap.
// ---------------------------------------------------------------------------
__global__ __launch_bounds__(256) void a1diff_kernel(
    const float* __restrict__ A0, const float* __restrict__ QKV,
    float* __restrict__ out0)
{
    const int L4 = LL / 4;
    size_t idx  = (size_t)blockIdx.x * 256 + threadIdx.x;  // over B*C*L*L/4
    size_t m4   = idx % L4;
    size_t rest = idx / L4;
    int l  = (int)(rest % LL);
    int bc = (int)(rest / LL);
    const float4* A0v = (const float4*)A0;
    float4* O = (float4*)out0;
    size_t rowBase = (size_t)bc * LL;
    float4 a0 = A0v[rowBase * L4 + m4];  // A0[b,c,0,:]
    float4 r;
    if (l < LL - 1) {
        float4 an = A0v[(rowBase + l + 1) * L4 + m4];
        float4 ac = A0v[(rowBase + l) * L4 + m4];
        r.x = an.x - a0.x - ac.x; r.y = an.y - a0.y - ac.y;
        r.z = an.z - a0.z - ac.z; r.w = an.w - a0.w - ac.w;
    } else {
        const float4* Qv = (const float4*)QKV;
        float4 q = Qv[(size_t)bc * L4 + m4];
        r.x = q.x - a0.x; r.y = q.y - a0.y;
        r.z = q.z - a0.z; r.w = q.w - a0.w;
    }
    O[(rowBase + l) * L4 + m4] = r;
}

// ---------------------------------------------------------------------------
// K7: main batched GEMM with fused diff:  out1 = D @ x0 - x0
//   D[l]   = A0[min(l+1,L-1)] - A0[l] + (l==L-1)*QKV   (branchless; diff folded)
//   Block tile 128x64 (256 thr = 8 waves); each wave computes a 16x64 strip ->
//   4 f32 WMMA accumulators, A fragment reused across 4 v_wmma (XDL-dense).
//   Double-buffered LDS (fetch k+1 regs -> compute k -> store -> 1 barrier).
//   LDS rows padded (20 / 68 floats): b128 stores stay 16B-aligned AND the
//   WMMA-layout reads are bank-conflict-free (stride 20 mod 64 -> 16 distinct).
// WMMA f32 16x16x4 lane layout:
//   A: lane&15 = M, half-waves hold K{0,1}/{2,3};  B mirrored on N;
//   C/D: VGPR v -> M = v (+8 for upper half-wave), lane&15 = N.
// ---------------------------------------------------------------------------
__global__ __launch_bounds__(256) void gemm_kernel(
    const float* __restrict__ A0, const float* __restrict__ x0,
    const float* __restrict__ QKV, float* __restrict__ out1)
{
    __shared__ float As[2][128][20];  // [buf][l-in-tile][k], padded row = 80B
    __shared__ float Bs[2][16][68];   // [buf][k][n-in-tile], padded row = 272B

    const int L4 = LL / 4, E4 = EE / 4;
    const int tiles_n = EE / 64;      // 8
    int bc       = blockIdx.y;
    int blockRow = (blockIdx.x / tiles_n) * 128;
    int blockCol = (blockIdx.x % tiles_n) * 64;
    int t = threadIdx.x;
    int lane = t & 31, wave = t >> 5;
    int kHi = lane >> 4;              // 0: lanes 0-15, 1: lanes 16-31
    int ln  = lane & 15;

    v8f acc[4];
    #pragma unroll
    for (int j = 0; j < 4; ++j) acc[j] = (v8f){0.f,0.f,0.f,0.f,0.f,0.f,0.f,0.f};

    const float4* Abase4 = (const float4*)A0 + (size_t)bc * LL * L4;
    const float4* Bbase4 = (const float4*)x0 + (size_t)bc * LL * E4;
    const float4* qrow4  = (const float4*)QKV + (size_t)bc * L4;

    // A-staging coords: 128x16 tile = 512 float4 -> 2 per thread
    int liA = t >> 2;                 // 0..63 (+64 on pass 1)
    int miq = t & 3;                  //  float4 column within 16-wide K tile
    int miA = miq * 4;
    // B-staging coords: 16x64 tile = 256 float4 -> 1 per thread
    int kiB = t >> 4;                 // 0..15
    int niB = (t & 15) * 4;
    int mRow = wave * 16 + ln;        // this wave's A row (per half-wave K pair)

    const int NT = LL / 16;           // 32 K-tiles

    float4 dA[2], dB;
    // ---- fetch K-tile kt into registers (branchless D-matrix build) ----
    auto fetch = [&](int kt, float4* fa, float4& fb) {
        int kq = (kt * 16) >> 2;
        #pragma unroll
        for (int p = 0; p < 2; ++p) {
            int l  = blockRow + liA + 64 * p;
            int l1 = (l + 1 < LL) ? (l + 1) : (LL - 1);
            float4 an = Abase4[(size_t)l1 * L4 + kq + miq];
            float4 ac = Abase4[(size_t)l  * L4 + kq + miq];
            float4 q  = qrow4[kq + miq];
            float isl = (l == LL - 1) ? 1.f : 0.f;
            fa[p].x = an.x - ac.x + isl * q.x;
            fa[p].y = an.y - ac.y + isl * q.y;
            fa[p].z = an.z - ac.z + isl * q.z;
            fa[p].w = an.w - ac.w + isl * q.w;
        }
        fb = Bbase4[(size_t)(kt * 16 + kiB) * E4 + (blockCol >> 2) + (t & 15)];
    };
    auto stage = [&](int buf, const float4* fa, const float4& fb) {
        *(float4*)&As[buf][liA][miA]      = fa[0];
        *(float4*)&As[buf][liA + 64][miA] = fa[1];
        *(float4*)&Bs[buf][kiB][niB]      = fb;
    };

    fetch(0, dA, dB);
    stage(0, dA, dB);
    __syncthreads();

    for (int kt = 0; kt < NT; ++kt) {
        int cur = kt & 1;
        bool hasNext = (kt + 1) < NT;
        float4 nA[2], nB;
        if (hasNext) fetch(kt + 1, nA, nB);   // global loads overlap WMMAs

        #pragma unroll
        for (int ks = 0; ks < 4; ++ks) {
            int k0 = ks * 4 + 2 * kHi;
            v2f a;
            a[0] = As[cur][mRow][k0];
            a[1] = As[cur][mRow][k0 + 1];
            #pragma unroll
            for (int j = 0; j < 4; ++j) {
                v2f b;
                b[0] = Bs[cur][k0][j * 16 + ln];
                b[1] = Bs[cur][k0 + 1][j * 16 + ln];
#if USE_WMMA_F32
                acc[j] = __builtin_amdgcn_wmma_f32_16x16x4_f32(
                    false, a, false, b, (short)0, acc[j], false, false);
#else
                // fallback: same C layout as the WMMA path
                #pragma unroll
                for (int v = 0; v < 8; ++v) {
                    float a0r = As[cur][wave * 16 + v + 8 * kHi][k0];
                    float a1r = As[cur][wave * 16 + v + 8 * kHi][k0 + 1];
                    acc[j][v] = fmaf(a0r, b[0], acc[j][v]);
                    acc[j][v] = fmaf(a1r, b[1], acc[j][v]);
                }
#endif
            }
        }
        if (hasNext) stage(1 - cur, nA, nB);  // write the other buffer
        __syncthreads();
    }

    // Epilogue: out1 = x1' - x0
    #pragma unroll
    for (int j = 0; j < 4; ++j) {
        int col = blockCol + j * 16 + ln;
        #pragma unroll
        for (int v = 0; v < 8; ++v) {
            int row = blockRow + wave * 16 + v + 8 * kHi;
            size_t idx = ((size_t)bc * LL + row) * EE + col;
            out1[idx] = acc[j][v] - x0[idx];
        }
    }
}

// ---------------------------------------------------------------------------
// K8: out2 = u1-u0 = w * (x1[L-1]/L - u0); x1[L-1] = (A0[L-1]-A0[0]+QKV) @ x0
// ---------------------------------------------------------------------------
__global__ __launch_bounds__(256) void u1_kernel(
    const float* __restrict__ A0, const float* __restrict__ x0,
    const float* __restrict__ u0, const float* __restrict__ QKV,
    const float* __restrict__ W, float* __restrict__ out2)
{
    __shared__ float coeff[LL];
    int bc = blockIdx.x, t = threadIdx.x;
    const float* Abase = A0 + (size_t)bc * LL * LL;
    for (int m = t; m < LL; m += 256)
        coeff[m] = Abase[(size_t)(LL - 1) * LL + m] - Abase[m] + QKV[(size_t)bc * LL + m];
    __syncthreads();
    float wv = W[bc];
    const float* Bbase = x0 + (size_t)bc * LL * EE;
    for (int e = t; e < EE; e += 256) {
        float acc = 0.f;
        for (int m = 0; m < LL; ++m)
            acc = fmaf(coeff[m], Bbase[(size_t)m * EE + e], acc);
        size_t o = (size_t)bc * EE + e;
        out2[o] = wv * (acc * (1.0f / LL) - u0[o]);
    }
}

// ---------------------------------------------------------------------------
extern "C" void kernel_launch(void* const* d_in, const int* in_sizes, int n_in,
                              void* d_out, int out_size, void* d_ws, size_t ws_size,
                              hipStream_t stream)
{
    const float* A0 = (const float*)d_in[0];
    const float* x0 = (const float*)d_in[1];
    const float* u0 = (const float*)d_in[2];
    const float* t0 = (const float*)d_in[3];
    const float* t1 = (const float*)d_in[4];
    const float* Qw = (const float*)d_in[5];
    const float* Qb = (const float*)d_in[6];
    const float* Vw = (const float*)d_in[7];
    const float* Vb = (const float*)d_in[8];
    const float* Kw = (const float*)d_in[9];
    const float* Kb = (const float*)d_in[10];
    const float* Tw = (const float*)d_in[11];
    // d_in[12] = T_bias: unused by the reference

    float* out0 = (float*)d_out;                         // (B,C,L,L)
    float* out1 = out0 + (size_t)BCT * LL * LL;          // (B,C,L,E)
    float* out2 = out1 + (size_t)BCT * LL * EE;          // (B,C,E)

    float* ws  = (float*)d_ws;                           // ~984 KB total
    float* T0  = ws;
    float* T1  = T0 + (size_t)BCT * EE;
    float* S   = T1 + (size_t)BCT * EE;
    float* G   = S  + (size_t)BCT * EE;
    float* QKV = G  + (size_t)BCT * EE;                  // logits -> softmax in-place
    float* W   = QKV + (size_t)BCT * LL;
    float* CB  = W + BCT;

    proj_kernel<<<dim3(EE / 128, BCT), 128, 0, stream>>>(
        x0, u0, t0, t1, Qw, Qb, Kw, Kb, Tw, T0, T1, S);
    wcb_kernel<<<BCT, 256, 0, stream>>>(T0, T1, S, Vb, W, CB);
    g_kernel<<<dim3(BCT, EE / 8), 256, 0, stream>>>(S, T1, Vw, G);
    logit_kernel<<<dim3(BCT, LL / 8), 256, 0, stream>>>(x0, G, CB, QKV);
    softmax_kernel<<<BCT, 256, 0, stream>>>(QKV);
    a1diff_kernel<<<(BCT * LL * (LL / 4)) / 256, 256, 0, stream>>>(A0, QKV, out0);
    gemm_kernel<<<dim3((LL / 128) * (EE / 64), BCT), 256, 0, stream>>>(A0, x0, QKV, out1);
    u1_kernel<<<BCT, 256, 0, stream>>>(A0, x0, u0, QKV, W, out2);
}